// LogMarginalLikelihood_60447369724423
// MI455X (gfx1250) — compile-verified
//
#include <hip/hip_runtime.h>
#include <math.h>

typedef __attribute__((ext_vector_type(2))) float v2f;
typedef __attribute__((ext_vector_type(8))) float v8f;

#define BLK 64
#define SOLVE_WAVES 16
#define TILE_LDS_FLOATS (BLK * (BLK + 2))   // [64][66] padded tile

// ---- CDNA5 Tensor Data Mover (TDM) support, probe-guarded -----------------
#if defined(__HIP_DEVICE_COMPILE__) &&                                   \
    __has_builtin(__builtin_amdgcn_tensor_load_to_lds) &&                \
    __has_builtin(__builtin_amdgcn_s_wait_tensorcnt)
#define USE_TDM 1
#else
#define USE_TDM 0
#endif

#if USE_TDM
typedef unsigned int u32x4 __attribute__((ext_vector_type(4)));
typedef int i32x4 __attribute__((ext_vector_type(4)));
typedef int i32x8 __attribute__((ext_vector_type(8)));

// DMA one BLKxBLK fp32 tile (row stride n elements) from global into LDS at
// byte offset lds_off, padding each 64-DWORD row by 2 DWORDs (-> [64][66]).
// D# layout per CDNA5 ISA (groups 0/1; 2D tensor).
__device__ __forceinline__ void tdm_load_tile(const float* gptr,
                                              unsigned lds_off, int n) {
  const unsigned long long ga = (unsigned long long)(uintptr_t)gptr;
  u32x4 g0;
  g0[0] = 1u;                                    // count=1, user descriptor
  g0[1] = lds_off;                               // lds_addr (bytes)
  g0[2] = (unsigned)(ga & 0xFFFFFFFFu);          // global_addr[31:0]
  g0[3] = (unsigned)((ga >> 32) & 0x01FFFFFFu)   // global_addr[56:32]
          | (2u << 30);                          // type = 2 ("image")
  i32x8 g1;
  // data_size=2 (4B), pad_enable, pad_interval=5 (64 DW), pad_amount=1 (2 DW)
  g1[0] = (int)((2u << 16) | (1u << 20) | (5u << 22) | (1u << 25));
  g1[1] = (int)(((unsigned)n & 0xFFFFu) << 16);            // tensor_dim0 lo16
  g1[2] = (int)((((unsigned)n >> 16) & 0xFFFFu)            // tensor_dim0 hi16
                | (((unsigned)n & 0xFFFFu) << 16));        // tensor_dim1 lo16
  g1[3] = (int)((((unsigned)n >> 16) & 0xFFFFu)            // tensor_dim1 hi16
                | ((unsigned)BLK << 16));                  // tile_dim0 = 64
  g1[4] = (int)BLK;                                        // tile_dim1 = 64
  g1[5] = n;                                               // dim0_stride lo32
  g1[6] = 0;
  g1[7] = 0;
  i32x4 z4 = {};
  i32x8 z8 = {};
  // 6-arg toolchain form: (g0, g1, g2, g3, g4, cpol)
  __builtin_amdgcn_tensor_load_to_lds(g0, g1, z4, z4, z8, 0);
}
#endif

// ---------------------------------------------------------------------------
// prep: KS = K + Sigma ; D[i][b] = mu[i] - Y[b*n + i] ; zero accumulators
// ---------------------------------------------------------------------------
__global__ void prep_kernel(const float* __restrict__ K,
                            const float* __restrict__ Sigma,
                            const float* __restrict__ mu,
                            const float* __restrict__ Y,
                            float* __restrict__ A, float* __restrict__ D,
                            float* __restrict__ scal, int n, int bs) {
  long long idx = (long long)blockIdx.x * blockDim.x + threadIdx.x;
  long long total = (long long)n * n;
  if (idx < total) A[idx] = K[idx] + Sigma[idx];
  if (idx < (long long)n * bs) {
    int i = (int)(idx / bs);
    int b = (int)(idx % bs);
    D[idx] = mu[i] - Y[(long long)b * n + i];
  }
  if (idx < 2) scal[idx] = 0.0f;
}

// ---------------------------------------------------------------------------
// Cholesky of the 64x64 diagonal block (one WG, 64 threads, LDS resident).
// Also accumulates logdet += 2*log(L_jj).
// ---------------------------------------------------------------------------
__global__ void chol_diag_kernel(float* __restrict__ A, float* __restrict__ scal,
                                 int n, int kb) {
  __shared__ float s[BLK][BLK + 1];
  const int tid = threadIdx.x;  // 0..63
  float* blk = A + (long long)(kb * BLK) * n + kb * BLK;
  for (int c = 0; c < BLK; ++c) s[tid][c] = blk[(long long)tid * n + c];
  __syncthreads();
  for (int j = 0; j < BLK; ++j) {
    if (tid == j) s[j][j] = sqrtf(s[j][j]);
    __syncthreads();
    float d = s[j][j];
    if (tid > j) s[tid][j] /= d;
    __syncthreads();
    if (tid > j) {
      float ljt = s[tid][j];
      for (int k2 = j + 1; k2 <= tid; ++k2) s[tid][k2] -= ljt * s[k2][j];
    }
    __syncthreads();
  }
  for (int c = 0; c < BLK; ++c) blk[(long long)tid * n + c] = s[tid][c];
  atomicAdd(&scal[0], 2.0f * logf(s[tid][tid]));
}

// ---------------------------------------------------------------------------
// Panel TRSM: B := B * L^{-T} for one 64x64 block-row below the diagonal.
// Row-parallel forward substitution, fully in LDS.
// ---------------------------------------------------------------------------
__global__ void trsm_kernel(float* __restrict__ A, int n, int kb) {
  __shared__ float Ld[BLK][BLK + 1];
  __shared__ float Bs[BLK][BLK + 1];
  const int tid = threadIdx.x;  // 0..63
  const int bi = kb + 1 + blockIdx.x;
  const float* L = A + (long long)(kb * BLK) * n + kb * BLK;
  float* B = A + (long long)(bi * BLK) * n + kb * BLK;
  for (int c = 0; c < BLK; ++c) {
    Ld[tid][c] = L[(long long)tid * n + c];
    Bs[tid][c] = B[(long long)tid * n + c];
  }
  __syncthreads();
  for (int j = 0; j < BLK; ++j) {
    float sacc = Bs[tid][j];
    for (int t = 0; t < j; ++t) sacc -= Bs[tid][t] * Ld[j][t];
    Bs[tid][j] = sacc / Ld[j][j];
  }
  __syncthreads();
  for (int c = 0; c < BLK; ++c) B[(long long)tid * n + c] = Bs[tid][c];
}

// ---------------------------------------------------------------------------
// SYRK trailing update: C(bi,bj) -= P(bi) * P(bj)^T  (64x64 tiles)
// Panels staged into DYNAMIC LDS by the Tensor Data Mover (TENSORcnt) when
// available (extern shared so the compiler cannot fold away the DMA-written
// reads), else by cooperative float4 loads. 8 waves x two 16x16 tiles,
// 16-step K loop of V_WMMA_F32_16X16X4_F32.
// ---------------------------------------------------------------------------
__global__ void syrk_kernel(float* __restrict__ A, int n, int kb) {
  extern __shared__ float smem[];  // 2 * TILE_LDS_FLOATS, dynamic => offset 0
  float (*Pi)[BLK + 2] = reinterpret_cast<float (*)[BLK + 2]>(smem);
  float (*Pj)[BLK + 2] =
      reinterpret_cast<float (*)[BLK + 2]>(smem + TILE_LDS_FLOATS);
  const int bj = kb + 1 + blockIdx.x;
  const int bi = kb + 1 + blockIdx.y;
  if (bj > bi) return;  // lower triangle only (uniform per-WG branch)
  const int tid = threadIdx.x;  // 0..255

#if USE_TDM
  if (tid < 32) {  // wave 0 issues both tile DMAs, then drains TENSORcnt
    const float* pi = A + (long long)(bi * BLK) * n + kb * BLK;
    const float* pj = A + (long long)(bj * BLK) * n + kb * BLK;
    tdm_load_tile(pi, 0u, n);
    tdm_load_tile(pj, (unsigned)(sizeof(float) * TILE_LDS_FLOATS), n);
    __builtin_amdgcn_s_wait_tensorcnt(0);
    asm volatile("" ::: "memory");  // DMA-written LDS: keep reads below
  }
  __syncthreads();
#else
  {  // cooperative stage: 4 threads per row, 16 floats each, float4 loads
    const int r = tid >> 2;
    const int c0 = (tid & 3) * 16;
    const float* pi = A + (long long)(bi * BLK) * n + kb * BLK;
    const float* pj = A + (long long)(bj * BLK) * n + kb * BLK;
    for (int c = 0; c < 16; c += 4) {
      const float4 vi = *(const float4*)(pi + (long long)r * n + c0 + c);
      const float4 vj = *(const float4*)(pj + (long long)r * n + c0 + c);
      Pi[r][c0 + c + 0] = vi.x; Pi[r][c0 + c + 1] = vi.y;
      Pi[r][c0 + c + 2] = vi.z; Pi[r][c0 + c + 3] = vi.w;
      Pj[r][c0 + c + 0] = vj.x; Pj[r][c0 + c + 1] = vj.y;
      Pj[r][c0 + c + 2] = vj.z; Pj[r][c0 + c + 3] = vj.w;
    }
  }
  __syncthreads();
#endif

  const int lane = tid & 31;
  const int wave = tid >> 5;
  const int t0 = wave * 2;           // two tiles per wave, same tile-row
  const int tm = t0 >> 2;
  const int tn0 = t0 & 3, tn1 = tn0 + 1;
  const int mrow = tm * 16 + (lane & 15);
  const int n0row = tn0 * 16 + (lane & 15);
  const int n1row = tn1 * 16 + (lane & 15);
  const int khalf = 2 * (lane >> 4);  // ISA 16x4 f32 A-layout: K = g + 2*(lane/16)

  v8f acc0 = {};
  v8f acc1 = {};
  for (int kk = 0; kk < 16; ++kk) {
    const int k0 = kk * 4 + khalf;
    v2f a;  a.x  = Pi[mrow][k0];  a.y  = Pi[mrow][k0 + 1];
    v2f b0; b0.x = Pj[n0row][k0]; b0.y = Pj[n0row][k0 + 1];
    v2f b1; b1.x = Pj[n1row][k0]; b1.y = Pj[n1row][k0 + 1];
    acc0 = __builtin_amdgcn_wmma_f32_16x16x4_f32(false, a, false, b0,
                                                 (short)0, acc0, false, false);
    acc1 = __builtin_amdgcn_wmma_f32_16x16x4_f32(false, a, false, b1,
                                                 (short)0, acc1, false, false);
  }

  // C/D layout: VGPR v -> row = tm*16 + v + 8*(lane>>4), col = tn*16 + (lane&15)
  float* C = A + (long long)(bi * BLK) * n + bj * BLK;
  const int crow = tm * 16 + 8 * (lane >> 4);
  const int ccol0 = tn0 * 16 + (lane & 15);
  const int ccol1 = tn1 * 16 + (lane & 15);
  for (int v = 0; v < 8; ++v) {
    C[(long long)(crow + v) * n + ccol0] -= acc0[v];
    C[(long long)(crow + v) * n + ccol1] -= acc1[v];
  }
}

// ---------------------------------------------------------------------------
// Blocked forward solve L Z = D for 32 RHS (D is n x 32, row-major), plus
// quad = sum(Z^2). One WG of 16 waves: the left-looking GEMM update is split
// across waves (WMMA f32 16x16x4), reduced in LDS; wave 0 solves the 16x16
// diagonal block serially and accumulates quad.
// ---------------------------------------------------------------------------
__global__ void solve_kernel(const float* __restrict__ A, float* __restrict__ D,
                             float* __restrict__ scal, int n, int bs) {
  __shared__ float red[SOLVE_WAVES][16 * 32];
  __shared__ float U[16 * 32];
  const int tid = threadIdx.x;
  const int lane = tid & 31;
  const int wave = tid >> 5;
  const int nb = n / 16;
  float qsum = 0.0f;

  const int mbase = (lane & 15);
  const int khalf = 2 * (lane >> 4);
  const int ncol0 = (lane & 15);
  const int ncol1 = 16 + (lane & 15);

  for (int ib = 0; ib < nb; ++ib) {
    // phase 1: partial update  U_partial = L[ib, 0:ib) @ Z[0:ib)
    v8f acc0 = {};
    v8f acc1 = {};
    const int mrow = ib * 16 + mbase;
    for (int c = wave; c < ib; c += SOLVE_WAVES) {
      const int kbase = c * 16;
      for (int ks = 0; ks < 4; ++ks) {
        const int k0 = kbase + ks * 4 + khalf;
        v2f a;  a.x  = A[(long long)mrow * n + k0];
                a.y  = A[(long long)mrow * n + k0 + 1];
        v2f b0; b0.x = D[(long long)k0 * bs + ncol0];
                b0.y = D[(long long)(k0 + 1) * bs + ncol0];
        v2f b1; b1.x = D[(long long)k0 * bs + ncol1];
                b1.y = D[(long long)(k0 + 1) * bs + ncol1];
        acc0 = __builtin_amdgcn_wmma_f32_16x16x4_f32(false, a, false, b0,
                                                     (short)0, acc0, false, false);
        acc1 = __builtin_amdgcn_wmma_f32_16x16x4_f32(false, a, false, b1,
                                                     (short)0, acc1, false, false);
      }
    }
    const int prow = 8 * (lane >> 4);
    const int pcol = (lane & 15);
    for (int v = 0; v < 8; ++v) {
      red[wave][(prow + v) * 32 + pcol]      = acc0[v];
      red[wave][(prow + v) * 32 + pcol + 16] = acc1[v];
    }
    __syncthreads();
    {  // reduction: 512 threads, one element each over 16 wave partials
      float ssum = 0.0f;
      for (int w = 0; w < SOLVE_WAVES; ++w) ssum += red[w][tid & 511];
      if (tid < 512) U[tid] = ssum;
    }
    __syncthreads();
    // phase 2: wave 0, lane b solves RHS column b through the diag block
    if (wave == 0) {
      const int b = lane;
      float z[16];
      const float* Ld = A + (long long)(ib * 16) * n + ib * 16;
      for (int r = 0; r < 16; ++r) {
        float zv = D[(long long)(ib * 16 + r) * bs + b] - U[r * 32 + b];
        for (int t = 0; t < r; ++t) zv -= Ld[(long long)r * n + t] * z[t];
        zv /= Ld[(long long)r * n + r];
        z[r] = zv;
        qsum += zv * zv;
        D[(long long)(ib * 16 + r) * bs + b] = zv;  // Z in place
      }
    }
    __threadfence();
    __syncthreads();
  }
  if (wave == 0) atomicAdd(&scal[1], qsum);
}

// ---------------------------------------------------------------------------
// final: out = quad/(2*bs) + 0.5*logdet + 0.5*tmax*log(2*pi)   ( = -lml )
// ---------------------------------------------------------------------------
__global__ void final_kernel(const float* __restrict__ scal,
                             const int* __restrict__ bsp,
                             const int* __restrict__ tmaxp,
                             float* __restrict__ out) {
  if (threadIdx.x == 0 && blockIdx.x == 0) {
    const float bsf = (float)bsp[0];
    const float tmaxf = (float)tmaxp[0];
    out[0] = scal[1] / (2.0f * bsf) + 0.5f * scal[0] +
             0.5f * tmaxf * logf(6.283185307179586f);
  }
}

extern "C" void kernel_launch(void* const* d_in, const int* in_sizes, int n_in,
                              void* d_out, int out_size, void* d_ws, size_t ws_size,
                              hipStream_t stream) {
  const float* Y     = (const float*)d_in[0];
  const float* mu    = (const float*)d_in[1];
  const float* K     = (const float*)d_in[2];
  const float* Sigma = (const float*)d_in[3];
  const int* bsp     = (const int*)d_in[4];
  const int* tmaxp   = (const int*)d_in[5];

  const int n = in_sizes[1];          // 4096
  const int bs = in_sizes[0] / n;     // 32

  float* A    = (float*)d_ws;                 // n*n fp32 (KS -> L)
  float* D    = A + (long long)n * n;         // n*bs fp32 (diff -> Z)
  float* scal = D + (long long)n * bs;        // [0]=logdet, [1]=quad

  {
    const long long total = (long long)n * n;
    const int threads = 256;
    const long long blocks = (total + threads - 1) / threads;
    prep_kernel<<<dim3((unsigned)blocks), threads, 0, stream>>>(
        K, Sigma, mu, Y, A, D, scal, n, bs);
  }

  const unsigned syrk_lds = 2u * TILE_LDS_FLOATS * sizeof(float);  // 33792 B
  const int NBb = n / BLK;  // 64 column blocks
  for (int kb = 0; kb < NBb; ++kb) {
    chol_diag_kernel<<<1, BLK, 0, stream>>>(A, scal, n, kb);
    const int rem = NBb - 1 - kb;
    if (rem > 0) {
      trsm_kernel<<<rem, BLK, 0, stream>>>(A, n, kb);
      syrk_kernel<<<dim3(rem, rem), 256, syrk_lds, stream>>>(A, n, kb);
    }
  }

  solve_kernel<<<1, 32 * SOLVE_WAVES, 0, stream>>>(A, D, scal, n, bs);
  final_kernel<<<1, 64, 0, stream>>>(scal, bsp, tmaxp, (float*)d_out);
}